// Conformer_11003706213223
// MI455X (gfx1250) — compile-verified
//
#include <hip/hip_runtime.h>

// ---------------- model constants ----------------
#define BATCH   8
#define NFEATS  128
#define T_IN    1600
#define DIMC    144
#define HEADS   4
#define DHH     36
#define FFD     576
#define NBLK    8
#define KDW     31
#define H1      63        // conv1 out height
#define W1      1598      // conv1 out width
#define TT      798       // time steps after conv2
#define FPRIME  31
#define FC1IN   (DIMC*FPRIME)   // 4464
#define ROWS    (BATCH*TT)      // 6384
#define NTOK    32
#define EPSN    1e-5f

typedef __attribute__((ext_vector_type(2))) float v2f;
typedef __attribute__((ext_vector_type(8))) float v8f;
typedef __attribute__((ext_vector_type(4))) unsigned int u32x4;
typedef __attribute__((ext_vector_type(8))) int i32x8;
typedef __attribute__((ext_vector_type(4))) int i32x4;

#if defined(__has_builtin)
#if __has_builtin(__builtin_amdgcn_tensor_load_to_lds) && __has_builtin(__builtin_amdgcn_s_wait_tensorcnt)
#define HAVE_TDM 1
#endif
#endif
#ifndef HAVE_TDM
#define HAVE_TDM 0
#endif

__device__ __forceinline__ float sigmoidf_(float x) { return 1.f / (1.f + __expf(-x)); }

#define WMMA_F32(a, b, c) __builtin_amdgcn_wmma_f32_16x16x4_f32(false, (a), false, (b), (short)0, (c), false, false)

// ---------------- generic batched f32 WMMA GEMM, 32x32 tile per wave ----------------
// C[b,h][m,n] = act( sum_k A[b,h][m,k] * B[b,h][k,n] + bias[n] )
// A element at A + b*sAb + h*sAh + m*aM + k      (A is ALWAYS k-contiguous, aK==1)
// B element at B + b*sBb + h*sBh + k*bK + n*bN
// C element at C + b*sCb + h*sCh + m*ldc + n
// Out-of-range rows/cols CLAMPED for addressing; garbage lanes never stored.
// MODE 0: strided B (bK arbitrary), K tail allowed (peeled, branchless selects)
// MODE 1: contiguous B (bK==1), K%4==0 -> all-b64 register path
// MODE 2: weight B (bN==1, K%16==0)   -> TDM-staged 16x32 LDS tile per block
struct GemmArgs {
  const float* __restrict__ A; const float* __restrict__ Bm;
  float* __restrict__ C; const float* __restrict__ bias;
  long long sAb, sAh, sBb, sBh, sCb, sCh;
  int aM, bK, bN, ldc;
  int M, N, K, act, H;
};

template<int MODE>
__global__ __launch_bounds__(128) void gemm_k(GemmArgs g) {
  const int wave = threadIdx.x >> 5, lane = threadIdx.x & 31;
  const int half = lane >> 4, r = lane & 15;
  const int m0 = (blockIdx.y * 4 + wave) * 32;
  const int n0 = blockIdx.x * 32;
  const int z  = blockIdx.z;
  const int bb = z / g.H, hh = z - bb * g.H;
  const float* A = g.A  + (long long)bb * g.sAb + (long long)hh * g.sAh;
  const float* B = g.Bm + (long long)bb * g.sBb + (long long)hh * g.sBh;
  float*       C = g.C  + (long long)bb * g.sCb + (long long)hh * g.sCh;

  const int Mm1 = g.M - 1, Nm1 = g.N - 1;
  const int ra0 = min(m0 + r,      Mm1);
  const int ra1 = min(m0 + 16 + r, Mm1);
  const int nb0 = min(n0 + r,      Nm1);
  const int nb1 = min(n0 + 16 + r, Nm1);

  // A fragment pointers: k-contiguous, 8B-aligned by construction (even strides/bases)
  const float* pa0 = A + (long long)ra0 * g.aM + 2 * half;
  const float* pa1 = A + (long long)ra1 * g.aM + 2 * half;

  v8f acc00 = {0.f,0.f,0.f,0.f,0.f,0.f,0.f,0.f};
  v8f acc01 = acc00, acc10 = acc00, acc11 = acc00;

  if constexpr (MODE == 2) {
#if HAVE_TDM
    __shared__ float tile[16 * 32];          // one 16(k) x 32(n) B tile per block
    for (int kk = 0; kk < g.K; kk += 16) {
      __syncthreads();                       // all waves done with previous tile
      if (threadIdx.x == 0) {
        // ---- build Tensor DMA Descriptor (D#) per CDNA5 ISA 8.3/8.4 ----
        unsigned long long ga = (unsigned long long)(size_t)(B + (long long)kk * g.bK + n0);
        unsigned lds = (unsigned)(size_t)&tile[0];          // low 32 bits = LDS offset
        u32x4 d0 = { 1u,                                    // count=1 (valid descriptor)
                     lds,                                   // lds_addr
                     (unsigned)ga,                          // global_addr[31:0]
                     (unsigned)((ga >> 32) & 0x1FFFFFFu) | 0x80000000u }; // ga[56:32] | type=2
        const unsigned td0 = (unsigned)(g.N - n0);          // tensor_dim0: n extent (HW zero-fills OOB)
        const unsigned td1 = (unsigned)(g.K - kk);          // tensor_dim1: k extent
        const unsigned long long st0 = (unsigned long long)g.bK;  // k-row stride (elements)
        i32x8 d1 = { (int)(2u << 16),                                   // data_size=4B
                     (int)((td0 & 0xFFFFu) << 16),                      // dim0[15:0] @ bits63:48
                     (int)(((td0 >> 16) & 0xFFFFu) | ((td1 & 0xFFFFu) << 16)),
                     (int)(((td1 >> 16) & 0xFFFFu) | (32u << 16)),      // tile_dim0 = 32 (n)
                     (int)16,                                           // tile_dim1 = 16 (k)
                     (int)(st0 & 0xFFFFFFFFull),                        // dim0_stride[31:0]
                     (int)((st0 >> 32) & 0xFFFFull),                    // dim0_stride[47:32]
                     0 };
        i32x4 dz4 = {0, 0, 0, 0};
        i32x8 dz8 = {0, 0, 0, 0, 0, 0, 0, 0};
        __builtin_amdgcn_tensor_load_to_lds(d0, d1, dz4, dz4, dz8, 0);
        __builtin_amdgcn_s_wait_tensorcnt(0);
      }
      __syncthreads();                       // tile visible to all waves
#pragma unroll
      for (int j = 0; j < 4; ++j) {
        const int kl = j * 4 + 2 * half;
        v2f a0 = *(const v2f*)pa0;
        v2f a1 = *(const v2f*)pa1;
        v2f b0, b1;
        b0.x = tile[kl * 32 + r];        b0.y = tile[(kl + 1) * 32 + r];
        b1.x = tile[kl * 32 + 16 + r];   b1.y = tile[(kl + 1) * 32 + 16 + r];
        acc00 = WMMA_F32(a0, b0, acc00);
        acc01 = WMMA_F32(a0, b1, acc01);
        acc10 = WMMA_F32(a1, b0, acc10);
        acc11 = WMMA_F32(a1, b1, acc11);
        pa0 += 4; pa1 += 4;
      }
    }
#else
    // fallback: register path with strided B (same semantics)
    const long long bK = g.bK;
    const float* pb0 = B + (long long)(2 * half) * bK + (long long)nb0;
    const float* pb1 = B + (long long)(2 * half) * bK + (long long)nb1;
    const long long dB = 4LL * bK;
#pragma unroll 2
    for (int kk = 0; kk < g.K; kk += 4) {
      v2f a0 = *(const v2f*)pa0, a1 = *(const v2f*)pa1;
      v2f b0, b1;
      b0.x = pb0[0]; b0.y = pb0[bK];
      b1.x = pb1[0]; b1.y = pb1[bK];
      acc00 = WMMA_F32(a0, b0, acc00);
      acc01 = WMMA_F32(a0, b1, acc01);
      acc10 = WMMA_F32(a1, b0, acc10);
      acc11 = WMMA_F32(a1, b1, acc11);
      pa0 += 4; pa1 += 4; pb0 += dB; pb1 += dB;
    }
#endif
  } else if constexpr (MODE == 1) {
    // bK == 1: both A and B fragments are contiguous b64 loads
    const float* pb0 = B + 2 * half + (long long)nb0 * g.bN;
    const float* pb1 = B + 2 * half + (long long)nb1 * g.bN;
#pragma unroll 2
    for (int kk = 0; kk < g.K; kk += 4) {
      v2f a0 = *(const v2f*)pa0, a1 = *(const v2f*)pa1;
      v2f b0 = *(const v2f*)pb0, b1 = *(const v2f*)pb1;
      acc00 = WMMA_F32(a0, b0, acc00);
      acc01 = WMMA_F32(a0, b1, acc01);
      acc10 = WMMA_F32(a1, b0, acc10);
      acc11 = WMMA_F32(a1, b1, acc11);
      pa0 += 4; pa1 += 4; pb0 += 4; pb1 += 4;
    }
  } else {
    // general strided B; guard-free main loop, peeled branchless K tail
    const long long bK = g.bK;
    const float* pb0 = B + (long long)(2 * half) * bK + (long long)nb0 * g.bN;
    const float* pb1 = B + (long long)(2 * half) * bK + (long long)nb1 * g.bN;
    const long long dB = 4LL * bK;
    const int K4 = g.K & ~3;
#pragma unroll 2
    for (int kk = 0; kk < K4; kk += 4) {
      v2f a0 = *(const v2f*)pa0, a1 = *(const v2f*)pa1;
      v2f b0, b1;
      b0.x = pb0[0]; b0.y = pb0[bK];
      b1.x = pb1[0]; b1.y = pb1[bK];
      acc00 = WMMA_F32(a0, b0, acc00);
      acc01 = WMMA_F32(a0, b1, acc01);
      acc10 = WMMA_F32(a1, b0, acc10);
      acc11 = WMMA_F32(a1, b1, acc11);
      pa0 += 4; pa1 += 4; pb0 += dB; pb1 += dB;
    }
    if (g.K & 3) {
      const int k0 = K4 + 2 * half;
      const bool u0 = k0 < g.K, u1 = k0 + 1 < g.K;
      v2f a0 = *(const v2f*)pa0, a1 = *(const v2f*)pa1;
      const float t00 = pb0[0], t01 = pb0[bK], t10 = pb1[0], t11 = pb1[bK];
      v2f b0, b1;
      a0.x = u0 ? a0.x : 0.f;  a0.y = u1 ? a0.y : 0.f;
      a1.x = u0 ? a1.x : 0.f;  a1.y = u1 ? a1.y : 0.f;
      b0.x = u0 ? t00 : 0.f;   b0.y = u1 ? t01 : 0.f;
      b1.x = u0 ? t10 : 0.f;   b1.y = u1 ? t11 : 0.f;
      acc00 = WMMA_F32(a0, b0, acc00);
      acc01 = WMMA_F32(a0, b1, acc01);
      acc10 = WMMA_F32(a1, b0, acc10);
      acc11 = WMMA_F32(a1, b1, acc11);
    }
  }

  const float bv0 = g.bias ? g.bias[nb0] : 0.f;
  const float bv1 = g.bias ? g.bias[nb1] : 0.f;
  const int c0 = n0 + r, c1 = n0 + 16 + r;
  const bool c0ok = c0 < g.N, c1ok = c1 < g.N;
#pragma unroll
  for (int i = 0; i < 8; ++i) {
    const int row0 = m0 + i + 8 * half;
    const int row1 = row0 + 16;
    if (row0 < g.M) {
      if (c0ok) { float v = acc00[i] + bv0; if (g.act == 1) v = v * sigmoidf_(v); C[(long long)row0 * g.ldc + c0] = v; }
      if (c1ok) { float v = acc01[i] + bv1; if (g.act == 1) v = v * sigmoidf_(v); C[(long long)row0 * g.ldc + c1] = v; }
    }
    if (row1 < g.M) {
      if (c0ok) { float v = acc10[i] + bv0; if (g.act == 1) v = v * sigmoidf_(v); C[(long long)row1 * g.ldc + c0] = v; }
      if (c1ok) { float v = acc11[i] + bv1; if (g.act == 1) v = v * sigmoidf_(v); C[(long long)row1 * g.ldc + c1] = v; }
    }
  }
}

// ---------------- conv1 (in-ch=1, 3x3, stride (2,1), VALID) + ReLU ----------------
__global__ __launch_bounds__(256) void conv1_relu_k(const float* __restrict__ spec,
    const float* __restrict__ w, const float* __restrict__ bias, float* __restrict__ out) {
  long long i = (long long)blockIdx.x * 256 + threadIdx.x;
  const long long n = (long long)BATCH * DIMC * H1 * W1;
  if (i >= n) return;
  int wp = (int)(i % W1); long long rr = i / W1;
  int h = (int)(rr % H1); rr /= H1;
  int c = (int)(rr % DIMC); int b = (int)(rr / DIMC);
  float s = bias[c];
#pragma unroll
  for (int kh = 0; kh < 3; ++kh)
#pragma unroll
    for (int kw = 0; kw < 3; ++kw)
      s += spec[((long long)b * NFEATS + (2*h + kh)) * T_IN + (wp + kw)] * w[c*9 + kh*3 + kw];
  out[i] = fmaxf(s, 0.f);
}

// ---------------- conv2 as implicit GEMM via WMMA f32, 32 rows x 16 cols per wave ----
__global__ __launch_bounds__(128) void conv2_wmma_k(const float* __restrict__ x1,
    const float* __restrict__ w2, const float* __restrict__ b2, float* __restrict__ out) {
  const int wave = threadIdx.x >> 5, lane = threadIdx.x & 31;
  const int half = lane >> 4, r = lane & 15;
  const int m0 = (blockIdx.y * 4 + wave) * 32;
  const int n0 = blockIdx.x * 16;
  const int f  = blockIdx.z;
  const int mA0 = min(m0 + r,      ROWS - 1);
  const int mA1 = min(m0 + 16 + r, ROWS - 1);
  const int bI0 = mA0 / TT, t0 = mA0 - bI0 * TT;
  const int bI1 = mA1 / TT, t1 = mA1 - bI1 * TT;
  const int c = n0 + r;                         // N = 144 exact
  const float* ab0 = x1 + (long long)bI0 * DIMC * H1 * W1 + 2 * t0;
  const float* ab1 = x1 + (long long)bI1 * DIMC * H1 * W1 + 2 * t1;
  const float* pw  = w2 + (long long)c * 1296 + 2 * half;

  v8f acc0 = {0.f,0.f,0.f,0.f,0.f,0.f,0.f,0.f};
  v8f acc1 = acc0;

  for (int kk = 0; kk < 1296; kk += 4) {
    const int k0 = kk + 2 * half;
    int ci = k0 / 9, rm = k0 - ci * 9, kh = rm / 3, kw = rm - kh * 3;
    const long long off0 = ((long long)ci * H1 + (2*f + kh)) * W1 + kw;
    const int k1 = k0 + 1;
    ci = k1 / 9; rm = k1 - ci * 9; kh = rm / 3; kw = rm - kh * 3;
    const long long off1 = ((long long)ci * H1 + (2*f + kh)) * W1 + kw;

    v2f a0, a1, b;
    a0.x = ab0[off0];  a0.y = ab0[off1];
    a1.x = ab1[off0];  a1.y = ab1[off1];
    b.x  = pw[kk];     b.y  = pw[kk + 1];
    acc0 = WMMA_F32(a0, b, acc0);
    acc1 = WMMA_F32(a1, b, acc1);
  }
  const float bias = b2[c];
#pragma unroll
  for (int i = 0; i < 8; ++i) {
    const int row0 = m0 + i + 8 * half;
    const int row1 = row0 + 16;
    if (row0 < ROWS) out[(long long)row0 * FC1IN + c * FPRIME + f] = fmaxf(acc0[i] + bias, 0.f);
    if (row1 < ROWS) out[(long long)row1 * FC1IN + c * FPRIME + f] = fmaxf(acc1[i] + bias, 0.f);
  }
}

// ---------------- LayerNorm over last dim (144), one wave per row ----------------
__global__ __launch_bounds__(32) void ln_k(const float* __restrict__ in, const float* __restrict__ g,
    const float* __restrict__ b, float* __restrict__ out) {
  int row = blockIdx.x;
  int lane = threadIdx.x;
  float vals[5];
  float s = 0.f;
#pragma unroll
  for (int j = 0; j < 5; ++j) {
    int c = lane + j * 32;
    float v = (c < DIMC) ? in[(long long)row * DIMC + c] : 0.f;
    vals[j] = v; s += v;
  }
  for (int o2 = 16; o2; o2 >>= 1) s += __shfl_xor(s, o2, 32);
  float mean = s * (1.f / DIMC);
  float vs = 0.f;
#pragma unroll
  for (int j = 0; j < 5; ++j) {
    int c = lane + j * 32;
    if (c < DIMC) { float d = vals[j] - mean; vs += d * d; }
  }
  for (int o2 = 16; o2; o2 >>= 1) vs += __shfl_xor(vs, o2, 32);
  float rs = rsqrtf(vs * (1.f / DIMC) + EPSN);
#pragma unroll
  for (int j = 0; j < 5; ++j) {
    int c = lane + j * 32;
    if (c < DIMC) out[(long long)row * DIMC + c] = (vals[j] - mean) * rs * g[c] + b[c];
  }
}

// ---------------- sinusoidal positional encoding ----------------
__global__ __launch_bounds__(256) void posenc_k(float* __restrict__ enc) {
  int i = blockIdx.x * 256 + threadIdx.x;
  if (i >= TT * DIMC) return;
  int t = i / DIMC, c = i - t * DIMC;
  int ii = c >> 1;
  float p = (float)t / __powf(10000.f, (2.f * (float)ii) / 144.f);
  enc[i] = (c & 1) ? __cosf(p) : __sinf(p);
}

// ---------------- elementwise helpers ----------------
__global__ __launch_bounds__(256) void axpy_k(float* __restrict__ dst, const float* __restrict__ src,
                                              float a, int n) {
  int i = blockIdx.x * 256 + threadIdx.x;
  if (i < n) dst[i] += a * src[i];
}

__global__ __launch_bounds__(256) void colbias_k(const float* __restrict__ in, const float* __restrict__ vec,
                                                 float* __restrict__ out, int n) {
  int i = blockIdx.x * 256 + threadIdx.x;
  if (i < n) out[i] = in[i] + vec[i % DIMC];
}

__global__ __launch_bounds__(256) void glu_k(const float* __restrict__ p1, float* __restrict__ out, int n) {
  int i = blockIdx.x * 256 + threadIdx.x;
  if (i >= n) return;
  int row = i / DIMC, c = i - row * DIMC;
  float a = p1[(long long)row * (2 * DIMC) + c];
  float g = p1[(long long)row * (2 * DIMC) + DIMC + c];
  out[i] = a * sigmoidf_(g);
}

// depthwise conv1d, SAME padding (K=31, pad 15), layout (B,T,C)
__global__ __launch_bounds__(256) void dwconv_k(const float* __restrict__ in, const float* __restrict__ w,
                                                const float* __restrict__ bias, float* __restrict__ out) {
  int i = blockIdx.x * 256 + threadIdx.x;
  if (i >= ROWS * DIMC) return;
  int rc = i / DIMC, c = i - rc * DIMC;
  int b = rc / TT, t = rc - b * TT;
  float s = bias[c];
#pragma unroll 4
  for (int k = 0; k < KDW; ++k) {
    int t2 = t + k - 15;
    if (t2 >= 0 && t2 < TT)
      s += in[((long long)b * TT + t2) * DIMC + c] * w[c * KDW + k];
  }
  out[i] = s;
}

// batchnorm stats per channel over (B,T)
__global__ __launch_bounds__(256) void bn_stats_k(const float* __restrict__ in,
                                                  float* __restrict__ mean, float* __restrict__ var) {
  int c = blockIdx.x;
  int tid = threadIdx.x, lane = tid & 31, wid = tid >> 5;
  float s = 0.f, s2 = 0.f;
  for (int i = tid; i < ROWS; i += 256) {
    float v = in[(long long)i * DIMC + c];
    s += v; s2 += v * v;
  }
  for (int o2 = 16; o2; o2 >>= 1) { s += __shfl_xor(s, o2, 32); s2 += __shfl_xor(s2, o2, 32); }
  __shared__ float rA[8], rB[8];
  if (lane == 0) { rA[wid] = s; rB[wid] = s2; }
  __syncthreads();
  if (tid == 0) {
    float S = 0.f, S2 = 0.f;
    for (int i = 0; i < 8; ++i) { S += rA[i]; S2 += rB[i]; }
    float m = S / (float)ROWS;
    mean[c] = m;
    var[c] = S2 / (float)ROWS - m * m;
  }
}

__global__ __launch_bounds__(256) void bn_apply_silu_k(float* __restrict__ t,
    const float* __restrict__ mean, const float* __restrict__ var,
    const float* __restrict__ g, const float* __restrict__ b, int n) {
  int i = blockIdx.x * 256 + threadIdx.x;
  if (i >= n) return;
  int c = i % DIMC;
  float v = (t[i] - mean[c]) * rsqrtf(var[c] + EPSN) * g[c] + b[c];
  t[i] = v * sigmoidf_(v);
}

// ---------------- fused rel-shift + scale + softmax over t (798) ----------------
__global__ __launch_bounds__(256) void attn_softmax_k(float* __restrict__ qk, const float* __restrict__ qp) {
  int row = blockIdx.x;                 // (b*HEADS + h)*TT + s
  int s = row % TT;
  int hb = row / TT;
  long long qkb = ((long long)hb * TT + s) * TT;
  long long qpb = (long long)hb * TT * TT;
  int tid = threadIdx.x, lane = tid & 31, wid = tid >> 5;

  float cache[4];
  float mx = -1e30f;
#pragma unroll
  for (int j = 0; j < 4; ++j) {
    int t = tid + j * 256;
    float v = -1e30f;
    if (t < TT) {
      int gidx = s * TT + t + TT;                 // rel-shift: f = a*S + b + S
      int ii = gidx / (TT + 1);
      int jj = gidx - ii * (TT + 1);
      float sh = (jj == 0) ? 0.f : qp[qpb + (long long)ii * TT + (jj - 1)];
      v = (qk[qkb + t] + sh) * (1.f / 12.f);      // 1/sqrt(144)
    }
    cache[j] = v;
    mx = fmaxf(mx, v);
  }
  __shared__ float redA[8], redB[8];
  for (int o2 = 16; o2; o2 >>= 1) mx = fmaxf(mx, __shfl_xor(mx, o2, 32));
  if (lane == 0) redA[wid] = mx;
  __syncthreads();
  float bm = redA[0];
#pragma unroll
  for (int i = 1; i < 8; ++i) bm = fmaxf(bm, redA[i]);

  float se = 0.f;
#pragma unroll
  for (int j = 0; j < 4; ++j) {
    int t = tid + j * 256;
    if (t < TT) { float e = __expf(cache[j] - bm); cache[j] = e; se += e; }
  }
  for (int o2 = 16; o2; o2 >>= 1) se += __shfl_xor(se, o2, 32);
  if (lane == 0) redB[wid] = se;
  __syncthreads();
  float bs = 0.f;
#pragma unroll
  for (int i = 0; i < 8; ++i) bs += redB[i];
  float inv = 1.f / bs;
#pragma unroll
  for (int j = 0; j < 4; ++j) {
    int t = tid + j * 256;
    if (t < TT) qk[qkb + t] = cache[j] * inv;
  }
}

// ---------------- log-softmax over 32 tokens, one wave per row ----------------
__global__ __launch_bounds__(32) void logsoftmax_k(const float* __restrict__ in, float* __restrict__ out) {
  int row = blockIdx.x, lane = threadIdx.x;
  float v = in[(long long)row * NTOK + lane];
  float m = v;
  for (int o2 = 16; o2; o2 >>= 1) m = fmaxf(m, __shfl_xor(m, o2, 32));
  float e = __expf(v - m);
  float s = e;
  for (int o2 = 16; o2; o2 >>= 1) s += __shfl_xor(s, o2, 32);
  out[(long long)row * NTOK + lane] = v - m - __logf(s);
}

__global__ void outlen_k(const int* __restrict__ slen, float* __restrict__ out) {
  int i = threadIdx.x;
  if (i < BATCH) {
    float f = ((float)slen[i] - 1.f) * 0.5f - 1.f;
    out[i] = (float)((int)f);
  }
}

// ---------------- host orchestration ----------------
static inline void gemm_launch(hipStream_t st,
    const float* A, long long sAb, long long sAh, int aM,
    const float* B, long long sBb, long long sBh, int bK, int bN,
    float* C, long long sCb, long long sCh, int ldc,
    const float* bias, int M, int N, int K, int act, int Bz, int Hz, int mode) {
  GemmArgs g;
  g.A = A; g.Bm = B; g.C = C; g.bias = bias;
  g.sAb = sAb; g.sAh = sAh; g.sBb = sBb; g.sBh = sBh; g.sCb = sCb; g.sCh = sCh;
  g.aM = aM; g.bK = bK; g.bN = bN; g.ldc = ldc;
  g.M = M; g.N = N; g.K = K; g.act = act; g.H = Hz;
  int mt32 = (M + 31) / 32;
  dim3 grid((N + 31) / 32, (mt32 + 3) / 4, Bz * Hz);
  if (mode == 2)      gemm_k<2><<<grid, dim3(128), 0, st>>>(g);
  else if (mode == 1) gemm_k<1><<<grid, dim3(128), 0, st>>>(g);
  else                gemm_k<0><<<grid, dim3(128), 0, st>>>(g);
}

static inline int ceilDiv(long long a, long long b) { return (int)((a + b - 1) / b); }

// input leaf order: setup_inputs() dict insertion order, depth-first through params/blocks
enum {
  I_SPEC = 0, I_SLEN, I_C1W, I_C1B, I_C2W, I_C2B, I_FC1W, I_FC1B, I_FCLW, I_FCLB,
  I_F1_LNG, I_F1_LNB, I_F1_W1, I_F1_B1, I_F1_W2, I_F1_B2,
  I_A_LNG, I_A_LNB, I_WQ, I_BQ, I_WK, I_BK, I_WV, I_BV, I_WO, I_BO, I_WP, I_U, I_V,
  I_C_LNG, I_C_LNB, I_PW1W, I_PW1B, I_DWW, I_DWB, I_BNG, I_BNB, I_PW2W, I_PW2B,
  I_F2_LNG, I_F2_LNB, I_F2_W1, I_F2_B1, I_F2_W2, I_F2_B2, I_LNG, I_LNB
};

extern "C" void kernel_launch(void* const* d_in, const int* in_sizes, int n_in,
                              void* d_out, int out_size, void* d_ws, size_t ws_size,
                              hipStream_t stream) {
  (void)in_sizes; (void)n_in; (void)out_size; (void)ws_size;
  auto F = [&](int i) { return (const float*)d_in[i]; };
  const float* spec = F(I_SPEC);
  const int*   slen = (const int*)d_in[I_SLEN];

  // ---- workspace layout (floats) ----
  float* W = (float*)d_ws;
  size_t off = 0;
  auto alloc = [&](size_t n) { float* p = W + off; off += n; return p; };
  float* X1  = alloc((size_t)BATCH * DIMC * H1 * W1);     // conv1 out, later reused for scores
  float* X2  = alloc((size_t)ROWS * FC1IN);               // fc1 input
  float* X   = alloc((size_t)ROWS * DIMC);                // main activation
  float* Y   = alloc((size_t)ROWS * DIMC);                // LN output / temp
  float* FFb = alloc((size_t)ROWS * FFD);
  float* Qb  = alloc((size_t)ROWS * DIMC);
  float* Kb  = alloc((size_t)ROWS * DIMC);
  float* Vb  = alloc((size_t)ROWS * DIMC);
  float* QU  = alloc((size_t)ROWS * DIMC);
  float* QV  = alloc((size_t)ROWS * DIMC);
  float* Ob  = alloc((size_t)ROWS * DIMC);
  float* P1  = alloc((size_t)ROWS * 2 * DIMC);            // pw1 out, reused for logits
  float* TD  = alloc((size_t)ROWS * DIMC);
  float* ENC = alloc((size_t)TT * DIMC);
  float* PB  = alloc((size_t)TT * DIMC);
  float* BNm = alloc(DIMC);
  float* BNv = alloc(DIMC);
  float* SC1 = X1;                                        // qk scores (B,H,S,S)
  float* SC2 = X1 + (size_t)BATCH * HEADS * TT * TT;      // qp scores
  float* LG  = P1;                                        // logits (ROWS,32)

  const int NELT = ROWS * DIMC;                           // 919,296
  const int EB = 256;

  // ---- subsampling front-end ----
  {
    long long n = (long long)BATCH * DIMC * H1 * W1;
    conv1_relu_k<<<ceilDiv(n, EB), EB, 0, stream>>>(spec, F(I_C1W), F(I_C1B), X1);
  }
  {
    int mt32 = (ROWS + 31) / 32;                          // 200
    dim3 grid(DIMC / 16, (mt32 + 3) / 4, FPRIME);         // (9, 50, 31)
    conv2_wmma_k<<<grid, dim3(128), 0, stream>>>(X1, F(I_C2W), F(I_C2B), X2);
  }
  gemm_launch(stream, X2, 0, 0, FC1IN,
                      F(I_FC1W), 0, 0, DIMC, 1,
                      X, 0, 0, DIMC, F(I_FC1B), ROWS, DIMC, FC1IN, 0, 1, 1, 2);
  posenc_k<<<ceilDiv((long long)TT * DIMC, EB), EB, 0, stream>>>(ENC);

  // ---- conformer blocks ----
  for (int blk = 0; blk < NBLK; ++blk) {
    const float* f1_lng = F(I_F1_LNG) + blk * DIMC;
    const float* f1_lnb = F(I_F1_LNB) + blk * DIMC;
    const float* f1_w1  = F(I_F1_W1)  + (size_t)blk * DIMC * FFD;
    const float* f1_b1  = F(I_F1_B1)  + blk * FFD;
    const float* f1_w2  = F(I_F1_W2)  + (size_t)blk * FFD * DIMC;
    const float* f1_b2  = F(I_F1_B2)  + blk * DIMC;
    const float* a_lng  = F(I_A_LNG)  + blk * DIMC;
    const float* a_lnb  = F(I_A_LNB)  + blk * DIMC;
    const float* wq = F(I_WQ) + (size_t)blk * DIMC * DIMC;  const float* bq = F(I_BQ) + blk * DIMC;
    const float* wk = F(I_WK) + (size_t)blk * DIMC * DIMC;  const float* bk = F(I_BK) + blk * DIMC;
    const float* wv = F(I_WV) + (size_t)blk * DIMC * DIMC;  const float* bv = F(I_BV) + blk * DIMC;
    const float* wo = F(I_WO) + (size_t)blk * DIMC * DIMC;  const float* bo = F(I_BO) + blk * DIMC;
    const float* wp = F(I_WP) + (size_t)blk * DIMC * DIMC;
    const float* up = F(I_U)  + blk * HEADS * DHH;
    const float* vp = F(I_V)  + blk * HEADS * DHH;
    const float* c_lng = F(I_C_LNG) + blk * DIMC;
    const float* c_lnb = F(I_C_LNB) + blk * DIMC;
    const float* pw1w = F(I_PW1W) + (size_t)blk * DIMC * 2 * DIMC;
    const float* pw1b = F(I_PW1B) + blk * 2 * DIMC;
    const float* dww  = F(I_DWW)  + (size_t)blk * DIMC * KDW;
    const float* dwb  = F(I_DWB)  + blk * DIMC;
    const float* bng  = F(I_BNG)  + blk * DIMC;
    const float* bnb  = F(I_BNB)  + blk * DIMC;
    const float* pw2w = F(I_PW2W) + (size_t)blk * DIMC * DIMC;
    const float* pw2b = F(I_PW2B) + blk * DIMC;
    const float* f2_lng = F(I_F2_LNG) + blk * DIMC;
    const float* f2_lnb = F(I_F2_LNB) + blk * DIMC;
    const float* f2_w1  = F(I_F2_W1)  + (size_t)blk * DIMC * FFD;
    const float* f2_b1  = F(I_F2_B1)  + blk * FFD;
    const float* f2_w2  = F(I_F2_W2)  + (size_t)blk * FFD * DIMC;
    const float* f2_b2  = F(I_F2_B2)  + blk * DIMC;
    const float* ln_g = F(I_LNG) + blk * DIMC;
    const float* ln_b = F(I_LNB) + blk * DIMC;

    // FFN1: x += 0.5 * (silu(ln(x)@w1+b1) @ w2 + b2)
    ln_k<<<ROWS, 32, 0, stream>>>(X, f1_lng, f1_lnb, Y);
    gemm_launch(stream, Y,0,0,DIMC, f1_w1,0,0,FFD,1, FFb,0,0,FFD, f1_b1, ROWS,FFD,DIMC, 1, 1,1, 2);
    gemm_launch(stream, FFb,0,0,FFD, f1_w2,0,0,DIMC,1, Ob,0,0,DIMC, f1_b2, ROWS,DIMC,FFD, 0, 1,1, 2);
    axpy_k<<<ceilDiv(NELT,EB), EB, 0, stream>>>(X, Ob, 0.5f, NELT);

    // attention
    ln_k<<<ROWS, 32, 0, stream>>>(X, a_lng, a_lnb, Y);
    gemm_launch(stream, Y,0,0,DIMC, wq,0,0,DIMC,1, Qb,0,0,DIMC, bq, ROWS,DIMC,DIMC, 0, 1,1, 2);
    gemm_launch(stream, Y,0,0,DIMC, wk,0,0,DIMC,1, Kb,0,0,DIMC, bk, ROWS,DIMC,DIMC, 0, 1,1, 2);
    gemm_launch(stream, Y,0,0,DIMC, wv,0,0,DIMC,1, Vb,0,0,DIMC, bv, ROWS,DIMC,DIMC, 0, 1,1, 2);
    colbias_k<<<ceilDiv(NELT,EB), EB, 0, stream>>>(Qb, up, QU, NELT);
    colbias_k<<<ceilDiv(NELT,EB), EB, 0, stream>>>(Qb, vp, QV, NELT);
    gemm_launch(stream, ENC,0,0,DIMC, wp,0,0,DIMC,1, PB,0,0,DIMC, nullptr, TT,DIMC,DIMC, 0, 1,1, 2);
    // qk[b,h,s,t] = sum_d QU[b,s,h,d] * K[b,t,h,d]   (batched over b,h; bK==1)
    gemm_launch(stream,
        QU,  (long long)TT*DIMC, DHH, DIMC,
        Kb,  (long long)TT*DIMC, DHH, 1, DIMC,
        SC1, (long long)HEADS*TT*TT, (long long)TT*TT, TT,
        nullptr, TT, TT, DHH, 0, BATCH, HEADS, 1);
    // qp[b,h,s,t] = sum_d QV[b,s,h,d] * P[t,h,d]
    gemm_launch(stream,
        QV,  (long long)TT*DIMC, DHH, DIMC,
        PB,  0, DHH, 1, DIMC,
        SC2, (long long)HEADS*TT*TT, (long long)TT*TT, TT,
        nullptr, TT, TT, DHH, 0, BATCH, HEADS, 1);
    attn_softmax_k<<<BATCH*HEADS*TT, 256, 0, stream>>>(SC1, SC2);
    // o[b,s,h,d] = sum_t att[b,h,s,t] * V[b,t,h,d]   (K=798 tail -> mode 0)
    gemm_launch(stream,
        SC1, (long long)HEADS*TT*TT, (long long)TT*TT, TT,
        Vb,  (long long)TT*DIMC, DHH, DIMC, 1,
        Ob,  (long long)TT*DIMC, DHH, DIMC,
        nullptr, TT, DHH, TT, 0, BATCH, HEADS, 0);
    gemm_launch(stream, Ob,0,0,DIMC, wo,0,0,DIMC,1, Y,0,0,DIMC, bo, ROWS,DIMC,DIMC, 0, 1,1, 2);
    axpy_k<<<ceilDiv(NELT,EB), EB, 0, stream>>>(X, Y, 1.0f, NELT);

    // conv module
    ln_k<<<ROWS, 32, 0, stream>>>(X, c_lng, c_lnb, Y);
    gemm_launch(stream, Y,0,0,DIMC, pw1w,0,0,2*DIMC,1, P1,0,0,2*DIMC, pw1b, ROWS,2*DIMC,DIMC, 0, 1,1, 2);
    glu_k<<<ceilDiv(NELT,EB), EB, 0, stream>>>(P1, Y, NELT);
    dwconv_k<<<ceilDiv(NELT,EB), EB, 0, stream>>>(Y, dww, dwb, TD);
    bn_stats_k<<<DIMC, 256, 0, stream>>>(TD, BNm, BNv);
    bn_apply_silu_k<<<ceilDiv(NELT,EB), EB, 0, stream>>>(TD, BNm, BNv, bng, bnb, NELT);
    gemm_launch(stream, TD,0,0,DIMC, pw2w,0,0,DIMC,1, Ob,0,0,DIMC, pw2b, ROWS,DIMC,DIMC, 0, 1,1, 2);
    axpy_k<<<ceilDiv(NELT,EB), EB, 0, stream>>>(X, Ob, 1.0f, NELT);

    // FFN2
    ln_k<<<ROWS, 32, 0, stream>>>(X, f2_lng, f2_lnb, Y);
    gemm_launch(stream, Y,0,0,DIMC, f2_w1,0,0,FFD,1, FFb,0,0,FFD, f2_b1, ROWS,FFD,DIMC, 1, 1,1, 2);
    gemm_launch(stream, FFb,0,0,FFD, f2_w2,0,0,DIMC,1, Ob,0,0,DIMC, f2_b2, ROWS,DIMC,FFD, 0, 1,1, 2);
    axpy_k<<<ceilDiv(NELT,EB), EB, 0, stream>>>(X, Ob, 0.5f, NELT);

    // final per-block LN (in place)
    ln_k<<<ROWS, 32, 0, stream>>>(X, ln_g, ln_b, X);
  }

  // ---- classifier head ----
  gemm_launch(stream, X,0,0,DIMC, F(I_FCLW),0,0,NTOK,1, LG,0,0,NTOK, F(I_FCLB), ROWS,NTOK,DIMC, 0, 1,1, 2);
  logsoftmax_k<<<ROWS, 32, 0, stream>>>(LG, (float*)d_out);
  outlen_k<<<1, 32, 0, stream>>>(slen, (float*)d_out + (size_t)ROWS * NTOK);
}